// GATLayer_12335146074235
// MI455X (gfx1250) — compile-verified
//
#include <hip/hip_runtime.h>
#include <math.h>

typedef __attribute__((ext_vector_type(2))) float v2f;
typedef __attribute__((ext_vector_type(8))) float v8f;

#define NEG_SLOPE 0.2f
#define BN_EPS 1e-5f

// LDS layout for W: K-pair interleaved, Wl[p][c][2] with p = k/2 (64 pairs,
// 256 cols). Pair stride padded 512 -> 544 dwords so the kg=1 lane half
// (offset 544 mod 64 = 32 banks) uses banks 32..63 while kg=0 uses 0..31:
// each B fragment is ONE conflict-free ds_load_b64 landing in adjacent VGPRs.
#define PAIR_STRIDE 544
#define ROWS_PER_BLOCK 256   // 8 row-tile iterations of 32 rows per block

// float atomic max via signed/unsigned int trick (correct for mixed signs, init = -inf)
__device__ __forceinline__ void atomic_max_f32(float* addr, float val) {
    if (val >= 0.0f)
        atomicMax((int*)addr, __float_as_int(val));
    else
        atomicMin((unsigned int*)addr, __float_as_uint(val));
}

// ---------------------------------------------------------------------------
// Kernel 0: initialize output accumulator, segment-max (-inf), denom, stats
// ---------------------------------------------------------------------------
__global__ void __launch_bounds__(256)
init_ws(float* __restrict__ out, float* __restrict__ m, float* __restrict__ denom,
        float* __restrict__ sums, long long nOut, int n4)
{
    long long gid = (long long)blockIdx.x * blockDim.x + threadIdx.x;
    long long stride = (long long)gridDim.x * blockDim.x;
    for (long long i = gid; i < nOut; i += stride) out[i] = 0.0f;
    for (long long i = gid; i < n4; i += stride) {
        m[i] = -__builtin_inff();
        denom[i] = 0.0f;
    }
    if (gid < 512) sums[gid] = 0.0f;   // sums[256] + sumsq[256], contiguous
}

// ---------------------------------------------------------------------------
// Kernel 1: h = x @ W via V_WMMA_F32_16X16X4_F32 (fp32, matches reference).
// Block = 256 threads = 8 waves. W (128x256) staged once per block into LDS
// (136 KB, K-pair interleaved), then the block sweeps 256 rows (8 iterations
// x 32 rows; wave w -> row tile (w>>2), col group (w&3)*64).
// Inner loop per K-step: one global b64 A load + 4 ds_load_b64 B fragments
// + 4 WMMAs, no repacking moves.
// ---------------------------------------------------------------------------
__global__ void __launch_bounds__(256)
gemm_proj(const float* __restrict__ X, const float* __restrict__ Wm,
          float* __restrict__ H, int n)
{
    extern __shared__ float Wl[];            // 64 * PAIR_STRIDE floats

    const int tid  = threadIdx.x;
    const int wave = tid >> 5;
    const int lane = tid & 31;

    // ---- cooperative stage of W into LDS, K-pair interleaved ----
    // For pair p (rows 2p, 2p+1) and col quad c4: read two float4s (one per
    // row), write four float2s {W[2p][c], W[2p+1][c]}.
    for (int i = tid; i < 4096; i += 256) {
        int p  = i >> 6;        // 0..63
        int c4 = (i & 63) * 4;  // 0..252
        float4 r0 = *(const float4*)(Wm + (size_t)(2 * p)     * 256 + c4);
        float4 r1 = *(const float4*)(Wm + (size_t)(2 * p + 1) * 256 + c4);
        float* dst = Wl + (size_t)p * PAIR_STRIDE + c4 * 2;
        ((v2f*)dst)[0] = (v2f){r0.x, r1.x};
        ((v2f*)dst)[1] = (v2f){r0.y, r1.y};
        ((v2f*)dst)[2] = (v2f){r0.z, r1.z};
        ((v2f*)dst)[3] = (v2f){r0.w, r1.w};
    }
    __syncthreads();

    const int mrow = lane & 15;     // A row / B,C column within tile
    const int kg   = lane >> 4;     // lane half selects K pair / C row half
    const int colBase = (wave & 3) * 64;
    // B fragment base: pair (k/2 + kg), column colBase+mrow
    const float* wlb = Wl + (size_t)kg * PAIR_STRIDE + (colBase + mrow) * 2;

    const int blockRow0 = blockIdx.x * ROWS_PER_BLOCK;

    #pragma unroll 1
    for (int rt = 0; rt < ROWS_PER_BLOCK / 32; ++rt) {
        const int rowBase = blockRow0 + rt * 32 + (wave >> 2) * 16;
        if (rowBase + 16 > n) continue;     // wave-uniform; N % 16 == 0 here

        v8f acc0 = {}; v8f acc1 = {}; v8f acc2 = {}; v8f acc3 = {};
        const float* xrow = X + (size_t)(rowBase + mrow) * 128 + 2 * kg;

        for (int k = 0; k < 128; k += 4) {
            // A 16x4 fragment: {X[row][k+2kg], X[row][k+2kg+1]}
            v2f a = *(const v2f*)(xrow + k);
            // B 4x16 fragments: single b64 per column tile
            const float* wp = wlb + (size_t)(k >> 1) * PAIR_STRIDE;
            v2f b0 = *(const v2f*)(wp);
            v2f b1 = *(const v2f*)(wp + 32);
            v2f b2 = *(const v2f*)(wp + 64);
            v2f b3 = *(const v2f*)(wp + 96);
            acc0 = __builtin_amdgcn_wmma_f32_16x16x4_f32(false, a, false, b0, (short)0, acc0, false, false);
            acc1 = __builtin_amdgcn_wmma_f32_16x16x4_f32(false, a, false, b1, (short)0, acc1, false, false);
            acc2 = __builtin_amdgcn_wmma_f32_16x16x4_f32(false, a, false, b2, (short)0, acc2, false, false);
            acc3 = __builtin_amdgcn_wmma_f32_16x16x4_f32(false, a, false, b3, (short)0, acc3, false, false);
        }

        // C/D layout: VGPR j holds row M = j + 8*(lane>>4), column N = lane&15
        const int r0 = rowBase + 8 * kg;
        float* hp = H + (size_t)r0 * 256 + colBase + mrow;
        #pragma unroll
        for (int j = 0; j < 8; ++j) {
            float* row = hp + (size_t)j * 256;
            row[0]  = acc0[j];
            row[16] = acc1[j];
            row[32] = acc2[j];
            row[48] = acc3[j];
        }
    }
}

// ---------------------------------------------------------------------------
// Kernel 2: a_src[n,h] = <h[n,h,:], att_src[h,:]>, same for a_dst.
// One thread per (node, head), float4 loads.
// ---------------------------------------------------------------------------
__global__ void __launch_bounds__(256)
att_reduce(const float* __restrict__ H, const float* __restrict__ attS,
           const float* __restrict__ attD, float* __restrict__ aS,
           float* __restrict__ aD, int n)
{
    int idx = blockIdx.x * 256 + threadIdx.x;
    if (idx >= n * 4) return;
    int node = idx >> 2, head = idx & 3;
    const float* hp = H + (size_t)node * 256 + head * 64;
    const float* sp = attS + head * 64;
    const float* dp = attD + head * 64;
    float s = 0.f, d = 0.f;
    #pragma unroll 4
    for (int c = 0; c < 64; c += 4) {
        float4 hv = *(const float4*)(hp + c);
        float4 sv = *(const float4*)(sp + c);
        float4 dv = *(const float4*)(dp + c);
        s += hv.x * sv.x + hv.y * sv.y + hv.z * sv.z + hv.w * sv.w;
        d += hv.x * dv.x + hv.y * dv.y + hv.z * dv.z + hv.w * dv.w;
    }
    aS[idx] = s;
    aD[idx] = d;
}

// ---------------------------------------------------------------------------
// Kernel 3: edge pass 1 — segment max of leaky_relu(a_src[src]+a_dst[dst]).
// One thread per edge (incl. self-loops), 4 heads each.
// ---------------------------------------------------------------------------
__global__ void __launch_bounds__(256)
edge_max(const int* __restrict__ ei, const float* __restrict__ aS,
         const float* __restrict__ aD, float* __restrict__ m, int E, int n)
{
    long long e = (long long)blockIdx.x * 256 + threadIdx.x;
    long long Et = (long long)E + n;
    if (e >= Et) return;
    int src, dst;
    if (e < E) { src = ei[e]; dst = ei[E + e]; }
    else       { src = dst = (int)(e - E); }
    const float4 s4 = *(const float4*)(aS + (size_t)src * 4);
    const float4 d4 = *(const float4*)(aD + (size_t)dst * 4);
    float t0 = s4.x + d4.x, t1 = s4.y + d4.y, t2 = s4.z + d4.z, t3 = s4.w + d4.w;
    t0 = t0 > 0.f ? t0 : NEG_SLOPE * t0;
    t1 = t1 > 0.f ? t1 : NEG_SLOPE * t1;
    t2 = t2 > 0.f ? t2 : NEG_SLOPE * t2;
    t3 = t3 > 0.f ? t3 : NEG_SLOPE * t3;
    float* mp = m + (size_t)dst * 4;
    atomic_max_f32(mp + 0, t0);
    atomic_max_f32(mp + 1, t1);
    atomic_max_f32(mp + 2, t2);
    atomic_max_f32(mp + 3, t3);
}

// ---------------------------------------------------------------------------
// Kernel 4: edge pass 2 — w = exp(e - m[dst]); denom[dst] += w;
// out[dst] += h[src] * w (unnormalized; divided by denom later).
// 64 threads per edge; thread j owns channels 4j..4j+3 (one head per group
// of 16 threads). Atomics hit L2 (h and out are both L2-resident on the
// 192 MB-L2 MI455X).
// ---------------------------------------------------------------------------
__global__ void __launch_bounds__(256)
edge_scatter(const int* __restrict__ ei, const float* __restrict__ H,
             const float* __restrict__ aS, const float* __restrict__ aD,
             const float* __restrict__ m, float* __restrict__ denom,
             float* __restrict__ out, int E, int n)
{
    long long gid = (long long)blockIdx.x * 256 + threadIdx.x;
    long long Et = (long long)E + n;
    long long e = gid >> 6;           // 64 threads per edge
    int j = (int)(gid & 63);
    if (e >= Et) return;
    int src, dst;
    if (e < E) { src = ei[e]; dst = ei[E + e]; }
    else       { src = dst = (int)(e - E); }

    const int c0 = j * 4;             // channels c0..c0+3 (single head)
    const int head = j >> 4;
    float t = aS[(size_t)src * 4 + head] + aD[(size_t)dst * 4 + head];
    float lrelu = t > 0.f ? t : NEG_SLOPE * t;
    float w = expf(lrelu - m[(size_t)dst * 4 + head]);
    if ((j & 15) == 0)
        atomicAdd(&denom[(size_t)dst * 4 + head], w);

    const float4 hv = *(const float4*)(H + (size_t)src * 256 + c0);
    float* op = out + (size_t)dst * 256 + c0;
    atomicAdd(op + 0, hv.x * w);
    atomicAdd(op + 1, hv.y * w);
    atomicAdd(op + 2, hv.z * w);
    atomicAdd(op + 3, hv.w * w);
}

// ---------------------------------------------------------------------------
// Kernel 5: normalize by denom, add bias, accumulate per-channel sum/sumsq.
// Thread t owns channel t; block handles a row slab (coalesced column reads).
// ---------------------------------------------------------------------------
__global__ void __launch_bounds__(256)
finalize_node(float* __restrict__ out, const float* __restrict__ denom,
              const float* __restrict__ bias, float* __restrict__ sums,
              float* __restrict__ sumsq, int n, int rowsPerBlock)
{
    const int c = threadIdx.x;     // 0..255
    const int head = c >> 6;
    const float b = bias[c];
    int r0 = blockIdx.x * rowsPerBlock;
    int r1 = min(n, r0 + rowsPerBlock);
    float s = 0.f, s2 = 0.f;
    for (int r = r0; r < r1; ++r) {
        float d = denom[(size_t)r * 4 + head] + 1e-16f;
        float v = out[(size_t)r * 256 + c] / d + b;
        out[(size_t)r * 256 + c] = v;
        s += v;
        s2 += v * v;
    }
    atomicAdd(&sums[c], s);
    atomicAdd(&sumsq[c], s2);
}

// ---------------------------------------------------------------------------
// Kernel 6: BatchNorm (batch stats, ddof=0) + ELU, in place on d_out.
// ---------------------------------------------------------------------------
__global__ void __launch_bounds__(256)
bn_elu(float* __restrict__ out, const float* __restrict__ sums,
       const float* __restrict__ sumsq, const float* __restrict__ gamma,
       const float* __restrict__ beta, int n)
{
    long long total = (long long)n * 256;
    long long stride = (long long)gridDim.x * 256;
    float invn = 1.0f / (float)n;
    for (long long i = (long long)blockIdx.x * 256 + threadIdx.x; i < total; i += stride) {
        int c = (int)(i & 255);
        float mean = sums[c] * invn;
        float var = fmaxf(sumsq[c] * invn - mean * mean, 0.0f);
        float v = out[i];
        float y = gamma[c] * (v - mean) * rsqrtf(var + BN_EPS) + beta[c];
        out[i] = y > 0.f ? y : expm1f(y);
    }
}

// ---------------------------------------------------------------------------
extern "C" void kernel_launch(void* const* d_in, const int* in_sizes, int n_in,
                              void* d_out, int out_size, void* d_ws, size_t ws_size,
                              hipStream_t stream)
{
    const float* x     = (const float*)d_in[0];
    const int*   ei    = (const int*)  d_in[1];   // [2,E] int32
    const float* W     = (const float*)d_in[2];   // [128,256]
    const float* attS  = (const float*)d_in[3];   // [4,64]
    const float* attD  = (const float*)d_in[4];
    const float* bias  = (const float*)d_in[5];
    const float* gamma = (const float*)d_in[6];
    const float* beta  = (const float*)d_in[7];
    float* out = (float*)d_out;

    const int n = in_sizes[0] / 128;
    const int E = in_sizes[1] / 2;

    // workspace layout (floats)
    float* ws   = (float*)d_ws;
    float* H    = ws;                       // n*256
    float* aS   = H  + (size_t)n * 256;     // n*4
    float* aD   = aS + (size_t)n * 4;       // n*4
    float* mArr = aD + (size_t)n * 4;       // n*4
    float* dArr = mArr + (size_t)n * 4;     // n*4
    float* sums = dArr + (size_t)n * 4;     // 256 + 256 (sumsq)

    const long long nOut = (long long)n * 256;
    init_ws<<<2048, 256, 0, stream>>>(out, mArr, dArr, sums, nOut, n * 4);

    const size_t wldsBytes = (size_t)64 * PAIR_STRIDE * sizeof(float);  // 136 KB
    gemm_proj<<<(n + ROWS_PER_BLOCK - 1) / ROWS_PER_BLOCK, 256, wldsBytes, stream>>>(
        x, W, H, n);

    att_reduce<<<(n * 4 + 255) / 256, 256, 0, stream>>>(H, attS, attD, aS, aD, n);

    const long long Et = (long long)E + n;
    edge_max<<<(unsigned)((Et + 255) / 256), 256, 0, stream>>>(ei, aS, aD, mArr, E, n);

    const long long work = Et * 64;
    edge_scatter<<<(unsigned)((work + 255) / 256), 256, 0, stream>>>(
        ei, H, aS, aD, mArr, dArr, out, E, n);

    const int rowsPerBlock = 256;
    finalize_node<<<(n + rowsPerBlock - 1) / rowsPerBlock, 256, 0, stream>>>(
        out, dArr, bias, sums, sums + 256, n, rowsPerBlock);

    bn_elu<<<2048, 256, 0, stream>>>(out, sums, sums + 256, gamma, beta, n);
}